// SpatialAttention_4337916969718
// MI455X (gfx1250) — compile-verified
//
#include <hip/hip_runtime.h>

typedef float v8f __attribute__((ext_vector_type(8)));
typedef float v2f __attribute__((ext_vector_type(2)));
typedef int   v4i __attribute__((ext_vector_type(4)));

#define Bq 4
#define Nq 1024
#define Hq 64
#define LPAD 68   // LDS row stride (floats): 272B = 16B-aligned rows, conflict-free

#if defined(__AMDGCN__) && __has_builtin(__builtin_amdgcn_global_load_async_to_lds_b128)
#define HAVE_ASYNC 1
typedef __attribute__((address_space(1))) v4i* gptr128;
typedef __attribute__((address_space(3))) v4i* lptr128;
#else
#define HAVE_ASYNC 0
#endif

__device__ __forceinline__ void async_tile_b128(const float* gsrc, float* ldst) {
#if HAVE_ASYNC
    __builtin_amdgcn_global_load_async_to_lds_b128(
        (gptr128)(void*)gsrc, (lptr128)(void*)ldst, 0, 0);
#else
    (void)gsrc; (void)ldst;
#endif
}

__device__ __forceinline__ void wait_async_le16() {
#if HAVE_ASYNC
#if __has_builtin(__builtin_amdgcn_s_wait_asynccnt)
    __builtin_amdgcn_s_wait_asynccnt(16);
#else
    asm volatile("s_wait_asynccnt 0x10" ::: "memory");
#endif
#endif
}
__device__ __forceinline__ void wait_async_le0() {
#if HAVE_ASYNC
#if __has_builtin(__builtin_amdgcn_s_wait_asynccnt)
    __builtin_amdgcn_s_wait_asynccnt(0);
#else
    asm volatile("s_wait_asynccnt 0x0" ::: "memory");
#endif
#endif
}

__device__ __forceinline__ float fast_tanh(float z) {
    // tanh(z) = 1 - 2/(exp(2z)+1); exp->v_exp_f32, rcp->v_rcp_f32
    float e = __expf(2.0f * z);
    return 1.0f - 2.0f * __builtin_amdgcn_rcpf(e + 1.0f);
}

// ---------------------------------------------------------------------------
// Kernel 1: s = x @ Ws^T + bs ; t = x @ Wt^T + bt  via f32 WMMA 16x16x4
// One wave handles 16 rows of the flattened (B*N, H) input.
// ---------------------------------------------------------------------------
__global__ __launch_bounds__(32)
void st_gemm_kernel(const float* __restrict__ x,
                    const float* __restrict__ Ws_w, const float* __restrict__ Ws_b,
                    const float* __restrict__ Wt_w, const float* __restrict__ Wt_b,
                    float* __restrict__ s_out, float* __restrict__ t_out)
{
    __shared__ __align__(16) float sX[16][LPAD];
    const int lane = threadIdx.x;
    const int m  = lane & 15;
    const int hi = lane >> 4;
    const int row0 = blockIdx.x * 16;            // flattened row over B*N

    for (int idx = lane; idx < 16 * Hq; idx += 32) {
        int r = idx >> 6, c = idx & 63;
        sX[r][c] = x[(row0 + r) * Hq + c];
    }
    __syncthreads();

    const v8f vz = {0.f,0.f,0.f,0.f,0.f,0.f,0.f,0.f};
    v8f accS[4], accT[4];
    #pragma unroll
    for (int c = 0; c < 4; ++c) { accS[c] = vz; accT[c] = vz; }

    #pragma unroll 4
    for (int kk = 0; kk < 16; ++kk) {
        const int kb = kk * 4;
        // A (16x4 f32): lane(hi,m) -> row m, K = kb+2*hi (+1 in .y)
        v2f a;
        a.x = sX[m][kb + 2 * hi];
        a.y = sX[m][kb + 2 * hi + 1];
        #pragma unroll
        for (int c = 0; c < 4; ++c) {
            const int ob = c * 16;
            // B (4x16): B[k][n] = W[n][k]; lane col n = m
            v2f bs, bt;
            bs.x = Ws_w[(ob + m) * Hq + kb + 2 * hi];
            bs.y = Ws_w[(ob + m) * Hq + kb + 2 * hi + 1];
            bt.x = Wt_w[(ob + m) * Hq + kb + 2 * hi];
            bt.y = Wt_w[(ob + m) * Hq + kb + 2 * hi + 1];
            accS[c] = __builtin_amdgcn_wmma_f32_16x16x4_f32(
                false, a, false, bs, (short)0, accS[c], false, false);
            accT[c] = __builtin_amdgcn_wmma_f32_16x16x4_f32(
                false, a, false, bt, (short)0, accT[c], false, false);
        }
    }

    // C layout: VGPR r, lane(hi,m) -> row M=r+8*hi, col N=m. Add bias, store.
    #pragma unroll
    for (int c = 0; c < 4; ++c) {
        const float bS = Ws_b[c * 16 + m];
        const float bT = Wt_b[c * 16 + m];
        #pragma unroll
        for (int r = 0; r < 8; ++r) {
            const int M = r + 8 * hi;
            s_out[(row0 + M) * Hq + c * 16 + m] = accS[c][r] + bS;
            t_out[(row0 + M) * Hq + c * 16 + m] = accT[c][r] + bT;
        }
    }
}

// ---------------------------------------------------------------------------
// Kernel 2: fused scores(tanh) + mask + online softmax + ctx WMMA + LayerNorm
// One wave per 16-row i-tile. Double-buffered j-tiles via async LDS loads.
// ---------------------------------------------------------------------------
__global__ __launch_bounds__(32)
void attn_fused_kernel(const float* __restrict__ x, const int* __restrict__ adj,
                       const float* __restrict__ s_in, const float* __restrict__ t_in,
                       const float* __restrict__ vvec,
                       const float* __restrict__ ln_g, const float* __restrict__ ln_b,
                       float* __restrict__ out)
{
    __shared__ __align__(16) float sS[16][LPAD];
    __shared__ __align__(16) float sT[2][16][LPAD];
    __shared__ __align__(16) float sX[2][16][LPAD];
    __shared__ float sP[16 * 17];
    __shared__ float sV[Hq];
    __shared__ float mrow[16], lrow[16];

    const int lane = threadIdx.x;
    const int m  = lane & 15;
    const int hi = lane >> 4;
    const int b  = blockIdx.x >> 6;      // N/16 = 64 tiles per batch
    const int i0 = (blockIdx.x & 63) * 16;
    const int rowbase = b * Nq + i0;

    // stage t/x j-tile `jt` into buffer `buf` (16 async b128 ops, or sync fallback)
    auto stage_tile = [&](int jt, int buf) {
        const int jbase = b * Nq + jt * 16;
#if HAVE_ASYNC
        const int row  = (lane >> 4);            // 2 rows per round
        const int col4 = (lane & 15) * 4;        // float4 column
        #pragma unroll
        for (int r2 = 0; r2 < 8; ++r2) {
            const int r = r2 * 2 + row;
            async_tile_b128(t_in + (jbase + r) * Hq + col4, &sT[buf][r][col4]);
            async_tile_b128(x    + (jbase + r) * Hq + col4, &sX[buf][r][col4]);
        }
#else
        for (int idx = lane; idx < 16 * Hq; idx += 32) {
            int r = idx >> 6, c = idx & 63;
            const int jr = (jbase + r) * Hq + c;
            sT[buf][r][c] = t_in[jr];
            sX[buf][r][c] = x[jr];
        }
#endif
    };

    stage_tile(0, 0);                            // prefetch first j-tile

    for (int idx = lane; idx < 16 * Hq; idx += 32) {
        int r = idx >> 6, c = idx & 63;
        sS[r][c] = s_in[(rowbase + r) * Hq + c];
    }
    sV[lane] = vvec[lane];
    sV[lane + 32] = vvec[lane + 32];
    if (lane < 16) { mrow[lane] = -3.0e38f; lrow[lane] = 0.0f; }
    __syncthreads();

    const v8f vz = {0.f,0.f,0.f,0.f,0.f,0.f,0.f,0.f};
    v8f acc[4];
    #pragma unroll
    for (int c = 0; c < 4; ++c) acc[c] = vz;

    for (int jt = 0; jt < Nq / 16; ++jt) {
        const int j0  = jt * 16;
        const int buf = jt & 1;
        if (jt + 1 < Nq / 16) {
            stage_tile(jt + 1, buf ^ 1);         // next tile in flight
            wait_async_le16();                   // current tile resident
        } else {
            wait_async_le0();
        }
        __syncthreads();

        // prefetch next adj tile rows (global_prefetch_b8)
        if (jt + 1 < Nq / 16)
            __builtin_prefetch(&adj[(b * Nq + i0 + m) * Nq + j0 + 16], 0, 0);

        float alpha_r[8];
        #pragma unroll
        for (int r = 0; r < 8; ++r) {
            const int M = r + 8 * hi;                   // C-layout row
            const float* srow = &sS[M][0];
            const float* trow = &sT[buf][m][0];         // col N=m -> t row j0+m
            float sc = 0.0f;
            #pragma unroll 8
            for (int h = 0; h < Hq; ++h) {
                sc = fmaf(fast_tanh(srow[h] + trow[h]), sV[h], sc);
            }
            const int ad = adj[(b * Nq + i0 + M) * Nq + j0 + m];
            if (ad == 0) sc = -1.0e9f;

            // row max over this 16-wide tile (within 16-lane half)
            float tmax = sc;
            tmax = fmaxf(tmax, __shfl_xor(tmax, 1, 32));
            tmax = fmaxf(tmax, __shfl_xor(tmax, 2, 32));
            tmax = fmaxf(tmax, __shfl_xor(tmax, 4, 32));
            tmax = fmaxf(tmax, __shfl_xor(tmax, 8, 32));

            const float mold = mrow[M];
            const float mnew = fmaxf(mold, tmax);
            const float al   = __expf(mold - mnew);
            const float pe   = __expf(sc - mnew);
            float rs = pe;
            rs += __shfl_xor(rs, 1, 32);
            rs += __shfl_xor(rs, 2, 32);
            rs += __shfl_xor(rs, 4, 32);
            rs += __shfl_xor(rs, 8, 32);
            if (m == 0) { mrow[M] = mnew; lrow[M] = lrow[M] * al + rs; }
            alpha_r[r] = al;
            sP[M * 17 + m] = pe;                        // stage P for A-layout
        }
        __syncthreads();

        // rescale running ctx accumulators by per-row alpha
        #pragma unroll
        for (int c = 0; c < 4; ++c)
            #pragma unroll
            for (int r = 0; r < 8; ++r)
                acc[c][r] *= alpha_r[r];

        // ctx(16xH) += P(16x16) @ Xj(16xH) ; K=16 in 4 WMMA steps, 4 H-chunks
        #pragma unroll
        for (int kk = 0; kk < 4; ++kk) {
            const int kb = kk * 4;
            v2f a;                                      // A-layout read of P
            a.x = sP[m * 17 + kb + 2 * hi];
            a.y = sP[m * 17 + kb + 2 * hi + 1];
            #pragma unroll
            for (int c = 0; c < 4; ++c) {
                v2f bb;                                 // B[k][n]=x[j0+k][c*16+n]
                bb.x = sX[buf][kb + 2 * hi][c * 16 + m];
                bb.y = sX[buf][kb + 2 * hi + 1][c * 16 + m];
                acc[c] = __builtin_amdgcn_wmma_f32_16x16x4_f32(
                    false, a, false, bb, (short)0, acc[c], false, false);
            }
        }
    }
    __syncthreads();

    // finalize: ctx /= l, then LayerNorm over H=64, write out
    #pragma unroll
    for (int r = 0; r < 8; ++r) {
        const int M = r + 8 * hi;
        const float linv = __builtin_amdgcn_rcpf(lrow[M]);
        float vr[4];
        float ssum = 0.0f;
        #pragma unroll
        for (int c = 0; c < 4; ++c) { vr[c] = acc[c][r] * linv; ssum += vr[c]; }
        ssum += __shfl_xor(ssum, 1, 32);
        ssum += __shfl_xor(ssum, 2, 32);
        ssum += __shfl_xor(ssum, 4, 32);
        ssum += __shfl_xor(ssum, 8, 32);
        const float mu = ssum * (1.0f / 64.0f);
        float vsum = 0.0f;
        #pragma unroll
        for (int c = 0; c < 4; ++c) { float d = vr[c] - mu; vsum += d * d; }
        vsum += __shfl_xor(vsum, 1, 32);
        vsum += __shfl_xor(vsum, 2, 32);
        vsum += __shfl_xor(vsum, 4, 32);
        vsum += __shfl_xor(vsum, 8, 32);
        const float rstd = rsqrtf(vsum * (1.0f / 64.0f) + 1e-5f);
        #pragma unroll
        for (int c = 0; c < 4; ++c) {
            const int h = c * 16 + m;
            out[(rowbase + M) * Hq + h] = (vr[c] - mu) * rstd * ln_g[h] + ln_b[h];
        }
    }
}

extern "C" void kernel_launch(void* const* d_in, const int* in_sizes, int n_in,
                              void* d_out, int out_size, void* d_ws, size_t ws_size,
                              hipStream_t stream) {
    (void)in_sizes; (void)n_in; (void)out_size; (void)ws_size;
    const float* x    = (const float*)d_in[0];
    const int*   adj  = (const int*)  d_in[1];
    const float* Ws_w = (const float*)d_in[2];
    const float* Ws_b = (const float*)d_in[3];
    const float* Wt_w = (const float*)d_in[4];
    const float* Wt_b = (const float*)d_in[5];
    const float* vvec = (const float*)d_in[6];
    const float* ln_g = (const float*)d_in[7];
    const float* ln_b = (const float*)d_in[8];
    float* out  = (float*)d_out;
    float* s_ws = (float*)d_ws;                 // B*N*H floats
    float* t_ws = s_ws + (size_t)Bq * Nq * Hq;  // B*N*H floats (2 MB total)

    st_gemm_kernel<<<(Bq * Nq) / 16, 32, 0, stream>>>(
        x, Ws_w, Ws_b, Wt_w, Wt_b, s_ws, t_ws);
    attn_fused_kernel<<<Bq * (Nq / 16), 32, 0, stream>>>(
        x, adj, s_ws, t_ws, vvec, ln_g, ln_b, out);
}